// AttentionModule_3719441679070
// MI455X (gfx1250) — compile-verified
//
#include <hip/hip_runtime.h>
#include <hip/hip_bf16.h>
#include <math.h>

typedef __attribute__((ext_vector_type(2))) float v2f;
typedef __attribute__((ext_vector_type(4))) float f4;
typedef __attribute__((ext_vector_type(8))) float v8f;

#define BATCH 64
#define DIM   16
#define NCH   4
#define IMG   224
#define NV    (IMG * IMG)     // 50176
#define VCH   5
#define HID   128
#define TEMP  10.0f
#define EPS   1e-5f

// ---------------------------------------------------------------------------
// Stage 1: 16-row MLP(16 -> 128 -leaky-> 16) + LayerNorm per wave, f32 WMMA.
// A-fragment layout (f32 16x16x4): lane holds row M=lane&15; VGPR j in lane
// half h holds K = 4*kc + 2*h + j.  C/D layout: lane=N (lane&15), VGPR v in
// half h holds row M = v + 8*h.
// ---------------------------------------------------------------------------
__device__ __forceinline__ void mlp16_ln(
    const v2f a[4],
    const float* __restrict__ W1, const float* __restrict__ b1,
    const float* __restrict__ W2, const float* __restrict__ b2,
    const float* __restrict__ g,  const float* __restrict__ bt,
    float* __restrict__ outRowMajor /*16x16*/, float* __restrict__ lds /*16x128*/)
{
    const int lane = threadIdx.x & 31;
    const int col  = lane & 15;
    const int hi   = lane >> 4;

    // ---- layer 1: X(16x16) @ W1(16x128) + b1, leaky ReLU -------------------
    v8f acc[8];
#pragma unroll
    for (int n = 0; n < 8; ++n) {
        float bias = b1[16 * n + col];
        v8f c;
#pragma unroll
        for (int v = 0; v < 8; ++v) c[v] = bias;
#pragma unroll
        for (int kc = 0; kc < 4; ++kc) {
            int k0 = 4 * kc + 2 * hi;
            v2f bf;
            bf.x = W1[(k0 + 0) * HID + 16 * n + col];
            bf.y = W1[(k0 + 1) * HID + 16 * n + col];
            c = __builtin_amdgcn_wmma_f32_16x16x4_f32(
                    false, a[kc], false, bf, (short)0, c, false, false);
        }
#pragma unroll
        for (int v = 0; v < 8; ++v) {
            float x = c[v];
            c[v] = x > 0.f ? x : 0.1f * x;       // leaky ReLU
        }
        acc[n] = c;
    }

    // ---- stage h1 (16x128) through LDS to build layer-2 A fragments --------
#pragma unroll
    for (int n = 0; n < 8; ++n)
#pragma unroll
        for (int v = 0; v < 8; ++v) {
            int M = v + 8 * hi;
            lds[M * HID + 16 * n + col] = acc[n][v];
        }
    __syncthreads();

    // ---- layer 2: h1(16x128) @ W2(128x16) + b2 -----------------------------
    float bias2 = b2[col];
    v8f c2;
#pragma unroll
    for (int v = 0; v < 8; ++v) c2[v] = bias2;
    const int row = lane & 15;
#pragma unroll
    for (int kc = 0; kc < 32; ++kc) {
        int k0 = 4 * kc + 2 * hi;
        v2f af, bf;
        af.x = lds[row * HID + k0 + 0];
        af.y = lds[row * HID + k0 + 1];
        bf.x = W2[(k0 + 0) * 16 + col];
        bf.y = W2[(k0 + 1) * 16 + col];
        c2 = __builtin_amdgcn_wmma_f32_16x16x4_f32(
                 false, af, false, bf, (short)0, c2, false, false);
    }

    // ---- LayerNorm over the 16 output features of each row -----------------
    float gc = g[col], bc = bt[col];
#pragma unroll
    for (int v = 0; v < 8; ++v) {
        float x = c2[v];
        float s = x;
#pragma unroll
        for (int m = 1; m < 16; m <<= 1) s += __shfl_xor(s, m, 32);
        float mu = s * (1.f / 16.f);
        float d  = x - mu;
        float vs = d * d;
#pragma unroll
        for (int m = 1; m < 16; m <<= 1) vs += __shfl_xor(vs, m, 32);
        float var = vs * (1.f / 16.f);
        float y = d * rsqrtf(var + EPS) * gc + bc;
        int M = v + 8 * hi;
        outRowMajor[M * 16 + col] = y;
    }
}

// blocks 0..3: q rows (64 batches); blocks 4..19: k rows (64 batches x 4 ch)
__global__ void __launch_bounds__(32)
mlp_embed_kernel(const float* __restrict__ q, const float* __restrict__ k,
                 const float* __restrict__ qW1, const float* __restrict__ qb1,
                 const float* __restrict__ qW2, const float* __restrict__ qb2,
                 const float* __restrict__ qg,  const float* __restrict__ qbt,
                 const float* __restrict__ kW1, const float* __restrict__ kb1,
                 const float* __restrict__ kW2, const float* __restrict__ kb2,
                 const float* __restrict__ kg,  const float* __restrict__ kbt,
                 float* __restrict__ qe, float* __restrict__ ke)
{
    __shared__ float lds[16 * HID];
    const int blk  = blockIdx.x;
    const int lane = threadIdx.x & 31;
    const int row  = lane & 15;
    const int hi   = lane >> 4;

    v2f a[4];
    if (blk < 4) {
        int gb = blk * 16 + row;                    // batch index
#pragma unroll
        for (int kc = 0; kc < 4; ++kc) {
            int d0 = 4 * kc + 2 * hi;
            a[kc].x = q[gb * DIM + d0 + 0];         // q[b, d, 0]
            a[kc].y = q[gb * DIM + d0 + 1];
        }
        mlp16_ln(a, qW1, qb1, qW2, qb2, qg, qbt, qe + blk * 256, lds);
    } else {
        int t = blk - 4;
        int r = t * 16 + row;                       // r = b*4 + c
        int b = r >> 2, c = r & 3;
#pragma unroll
        for (int kc = 0; kc < 4; ++kc) {
            int d0 = 4 * kc + 2 * hi;
            a[kc].x = k[b * 64 + (d0 + 0) * NCH + c];   // k[b, d, 0, c]
            a[kc].y = k[b * 64 + (d0 + 1) * NCH + c];
        }
        mlp16_ln(a, kW1, kb1, kW2, kb2, kg, kbt, ke + t * 256, lds);
    }
}

// ---------------------------------------------------------------------------
// Stage 2: per-batch dot products + both softmaxes (64 batches, trivial).
// ---------------------------------------------------------------------------
__global__ void __launch_bounds__(64)
small_attn_kernel(const float* __restrict__ qe, const float* __restrict__ ke,
                  const float* __restrict__ drop,
                  float* __restrict__ att, float* __restrict__ cw_out)
{
    int b = blockIdx.x * blockDim.x + threadIdx.x;
    if (b >= BATCH) return;

    float dot[NCH];
#pragma unroll
    for (int c = 0; c < NCH; ++c) {
        float s = 0.f;
#pragma unroll
        for (int d = 0; d < DIM; ++d)
            s += qe[b * DIM + d] * ke[(b * NCH + c) * DIM + d];
        dot[c] = s * 0.25f;                          // * dim^-0.5
    }

    // contrastwise = softmax(dot / TEMP)
    float m1 = -1e30f;
#pragma unroll
    for (int c = 0; c < NCH; ++c) m1 = fmaxf(m1, dot[c] / TEMP);
    float e[NCH], se = 0.f;
#pragma unroll
    for (int c = 0; c < NCH; ++c) { e[c] = __expf(dot[c] / TEMP - m1); se += e[c]; }
#pragma unroll
    for (int c = 0; c < NCH; ++c) cw_out[b * NCH + c] = e[c] / se;

    // att = softmax((dot - 1e5*dropout) / TEMP)   (pixel-independent)
    float lg[NCH];
    float m2 = -1e30f;
#pragma unroll
    for (int c = 0; c < NCH; ++c) {
        lg[c] = (dot[c] - 100000.0f * drop[b * NCH + c]) / TEMP;
        m2 = fmaxf(m2, lg[c]);
    }
    float se2 = 0.f;
#pragma unroll
    for (int c = 0; c < NCH; ++c) { lg[c] = __expf(lg[c] - m2); se2 += lg[c]; }
#pragma unroll
    for (int c = 0; c < NCH; ++c) att[b * NCH + c] = lg[c] / se2;
}

// ---------------------------------------------------------------------------
// Stage 3: bandwidth-bound streaming pass (~424 MB total, NT hints).
// grid = (NV/256, BATCH), block = 256.
// ---------------------------------------------------------------------------
__global__ void __launch_bounds__(256)
stream_kernel(const float* __restrict__ v, const float* __restrict__ mask,
              const float* __restrict__ att,
              float* __restrict__ vout, float* __restrict__ aout)
{
    const int b = blockIdx.y;
    const int p = blockIdx.x * blockDim.x + threadIdx.x;   // pixel id < NV

    // att[b, 0..3] (hot in cache, shared by whole batch)
    f4 a4 = *(const f4*)(att + b * NCH);

    // mask[b, 0, y, x, 0..3] — contiguous 16B, streamed once
    f4 m4 = __builtin_nontemporal_load((const f4*)(mask + ((size_t)b * NV + p) * NCH));

    float n0 = a4.x * m4.x;
    float n1 = a4.y * m4.y;
    float n2 = a4.z * m4.z;
    float n3 = a4.w * m4.w;
    float inv = 1.0f / (n0 + n1 + n2 + n3 + EPS);
    n0 *= inv; n1 *= inv; n2 *= inv; n3 *= inv;

    // aout[b, c, y, x]
    size_t ao = (size_t)b * NCH * NV + p;
    __builtin_nontemporal_store(n0, aout + ao + 0 * (size_t)NV);
    __builtin_nontemporal_store(n1, aout + ao + 1 * (size_t)NV);
    __builtin_nontemporal_store(n2, aout + ao + 2 * (size_t)NV);
    __builtin_nontemporal_store(n3, aout + ao + 3 * (size_t)NV);

    // vout[b, vch, y, x] = sum_c norm[c] * v[b, vch, p, c]
    size_t vbase = ((size_t)b * VCH) * NV + p;
#pragma unroll
    for (int vc = 0; vc < VCH; ++vc) {
        f4 vv = __builtin_nontemporal_load(
                    (const f4*)(v + (vbase + (size_t)vc * NV) * NCH));
        float o = n0 * vv.x + n1 * vv.y + n2 * vv.z + n3 * vv.w;
        __builtin_nontemporal_store(o, vout + vbase + (size_t)vc * NV);
    }
}

// ---------------------------------------------------------------------------
extern "C" void kernel_launch(void* const* d_in, const int* in_sizes, int n_in,
                              void* d_out, int out_size, void* d_ws, size_t ws_size,
                              hipStream_t stream)
{
    const float* q    = (const float*)d_in[0];
    const float* k    = (const float*)d_in[1];
    const float* v    = (const float*)d_in[2];
    const float* mask = (const float*)d_in[3];
    const float* drop = (const float*)d_in[4];
    const float* qW1  = (const float*)d_in[5];
    const float* qb1  = (const float*)d_in[6];
    const float* qW2  = (const float*)d_in[7];
    const float* qb2  = (const float*)d_in[8];
    const float* qg   = (const float*)d_in[9];
    const float* qbt  = (const float*)d_in[10];
    const float* kW1  = (const float*)d_in[11];
    const float* kb1  = (const float*)d_in[12];
    const float* kW2  = (const float*)d_in[13];
    const float* kb2  = (const float*)d_in[14];
    const float* kg   = (const float*)d_in[15];
    const float* kbt  = (const float*)d_in[16];

    float* out = (float*)d_out;
    const size_t VOUT_SZ = (size_t)BATCH * VCH * NV;   // 16,056,320
    const size_t AOUT_SZ = (size_t)BATCH * NCH * NV;   // 12,845,056
    float* vout = out;
    float* aout = out + VOUT_SZ;
    float* cw   = out + VOUT_SZ + AOUT_SZ;             // 256 floats

    float* ws  = (float*)d_ws;
    float* qe  = ws;                   //  64*16 = 1024 floats
    float* ke  = ws + 1024;            // 256*16 = 4096 floats
    float* att = ws + 1024 + 4096;     //  64*4  =  256 floats

    // Stage 1: WMMA MLP+LN  (4 q-tiles + 16 k-tiles, one wave each)
    mlp_embed_kernel<<<20, 32, 0, stream>>>(q, k,
        qW1, qb1, qW2, qb2, qg, qbt,
        kW1, kb1, kW2, kb2, kg, kbt, qe, ke);

    // Stage 2: per-batch softmaxes
    small_attn_kernel<<<1, 64, 0, stream>>>(qe, ke, drop, att, cw);

    // Stage 3: bandwidth-bound streaming pass
    dim3 grid(NV / 256, BATCH);
    stream_kernel<<<grid, 256, 0, stream>>>(v, mask, att, vout, aout);
}